// HANGAE_17798344474650
// MI455X (gfx1250) — compile-verified
//
#include <hip/hip_runtime.h>
#include <cstddef>

#define Nn 50000
#define Ecnt 800000
#define INDIM 128
#define ACT_NONE 0
#define ACT_ELU  1
#define ACT_TANH 2

typedef float v2f __attribute__((ext_vector_type(2)));
typedef float v8f __attribute__((ext_vector_type(8)));

// ---------------- WMMA GEMM: C[M,Nc] = A[M,K] @ B[K,Nc] (+bias, +act) ------
// Row-major everywhere. K % 4 == 0, Nc % 64 == 0. One wave computes a 16x64 tile.
// Partial M-tile: A row index is clamped (WMMA output rows are independent, so
// garbage rows are harmless); only the stores are row-predicated. This keeps
// the inner loop free of EXEC manipulation (full-EXEC requirement for WMMA).
template<int ACT>
__global__ void k_gemm_wmma(const float* __restrict__ A, const float* __restrict__ B,
                            const float* __restrict__ bias, float* __restrict__ C,
                            int M, int K, int Nc) {
  const int lane = threadIdx.x & 31;
  const int wave = blockIdx.x * (blockDim.x >> 5) + (threadIdx.x >> 5);
  const int ntn = Nc >> 6;
  const int mtiles = (M + 15) >> 4;
  const int tm = wave / ntn;
  if (tm >= mtiles) return;                    // wave-uniform exit (EXEC stays full)
  const int nb = (wave - tm * ntn) << 6;

  const int row  = tm * 16 + (lane & 15);
  const int khi  = (lane >> 4) << 1;           // lanes 16-31 hold K+2,K+3
  const int rowc = row < M ? row : M - 1;      // clamp: no predication in loop
  const int col  = nb + (lane & 15);

  const float* Ap = A + (size_t)rowc * K + khi;          // 8B-aligned (khi even)
  const float* Bp = B + (size_t)khi * Nc + col;

  v8f acc0 = {}, acc1 = {}, acc2 = {}, acc3 = {};
  #pragma unroll 2
  for (int k = 0; k < K; k += 4) {
    v2f a = *(const v2f*)Ap;                   // one global_load_b64
    Ap += 4;
    v2f b0, b1, b2, b3;
    b0.x = Bp[0];  b0.y = Bp[Nc];
    b1.x = Bp[16]; b1.y = Bp[Nc + 16];
    b2.x = Bp[32]; b2.y = Bp[Nc + 32];
    b3.x = Bp[48]; b3.y = Bp[Nc + 48];
    Bp += (size_t)4 * Nc;
    acc0 = __builtin_amdgcn_wmma_f32_16x16x4_f32(false, a, false, b0, (short)0, acc0, false, false);
    acc1 = __builtin_amdgcn_wmma_f32_16x16x4_f32(false, a, false, b1, (short)0, acc1, false, false);
    acc2 = __builtin_amdgcn_wmma_f32_16x16x4_f32(false, a, false, b2, (short)0, acc2, false, false);
    acc3 = __builtin_amdgcn_wmma_f32_16x16x4_f32(false, a, false, b3, (short)0, acc3, false, false);
  }
  const int r0 = tm * 16 + ((lane >> 4) << 3);   // C/D: VGPR r = rows r, r+8
  #pragma unroll
  for (int r = 0; r < 8; ++r) {
    int orow = r0 + r;
    if (orow < M) {
      float vv[4] = { acc0[r], acc1[r], acc2[r], acc3[r] };
      float* Cp = C + (size_t)orow * Nc + col;
      #pragma unroll
      for (int j = 0; j < 4; ++j) {
        float v = vv[j];
        if (bias) v += bias[col + 16 * j];
        if (ACT == ACT_ELU)  v = v > 0.f ? v : expm1f(v);
        if (ACT == ACT_TANH) v = tanhf(v);
        Cp[16 * j] = v;
      }
    }
  }
}

// ---------------- utility kernels ------------------------------------------
__global__ void k_zero(float* __restrict__ p, size_t n) {
  size_t i = (size_t)blockIdx.x * blockDim.x + threadIdx.x;
  if (i < n) p[i] = 0.f;
}

__global__ void k_transpose64(const float* __restrict__ W, float* __restrict__ WT) {
  int t = blockIdx.x * blockDim.x + threadIdx.x;
  if (t >= 64 * 64) return;
  int j = t >> 6, k = t & 63;
  WT[k * 64 + j] = W[j * 64 + k];
}

// a_src[n,h] = sum_c h[n,h,c]*att_src[h,c]  (h stored [N, 4*64])
__global__ void k_attn_proj(const float* __restrict__ h, const float* __restrict__ att_src,
                            const float* __restrict__ att_dst, float* __restrict__ asrc,
                            float* __restrict__ adst) {
  int t = blockIdx.x * blockDim.x + threadIdx.x;
  if (t >= Nn * 4) return;
  int n = t >> 2, hh = t & 3;
  const float* hp = h + (size_t)n * 256 + hh * 64;
  const float* as = att_src + hh * 64;
  const float* ad = att_dst + hh * 64;
  float s = 0.f, d = 0.f;
  #pragma unroll 8
  for (int c = 0; c < 64; ++c) { float v = hp[c]; s += v * as[c]; d += v * ad[c]; }
  asrc[t] = s; adst[t] = d;
}

// monotone float <-> uint for atomicMax over signed floats
__device__ __forceinline__ unsigned f2ord(float f) {
  unsigned u = __float_as_uint(f);
  return (u & 0x80000000u) ? ~u : (u | 0x80000000u);
}
__device__ __forceinline__ float ord2f(unsigned u) {
  return (u & 0x80000000u) ? __uint_as_float(u & 0x7fffffffu) : __uint_as_float(~u);
}

__device__ __forceinline__ void edge_sd(const int* __restrict__ ei, int eid, int& s, int& d) {
  if (eid < Ecnt) { s = ei[eid]; d = ei[Ecnt + eid]; }
  else            { s = d = eid - Ecnt; }        // appended self loops
}
__device__ __forceinline__ float edge_e(const float* asrc, const float* adst, int s, int d, int hh) {
  float e = asrc[s * 4 + hh] + adst[d * 4 + hh];
  return e > 0.f ? e : 0.2f * e;                 // leaky_relu 0.2
}

__global__ void k_edge_max(const int* __restrict__ ei, const float* __restrict__ asrc,
                           const float* __restrict__ adst, unsigned* __restrict__ mkey) {
  int t = blockIdx.x * blockDim.x + threadIdx.x;
  if (t >= (Ecnt + Nn) * 4) return;
  int eid = t >> 2, hh = t & 3, s, d;
  edge_sd(ei, eid, s, d);
  atomicMax(&mkey[d * 4 + hh], f2ord(edge_e(asrc, adst, s, d, hh)));
}

__global__ void k_edge_den(const int* __restrict__ ei, const float* __restrict__ asrc,
                           const float* __restrict__ adst, const unsigned* __restrict__ mkey,
                           float* __restrict__ den) {
  int t = blockIdx.x * blockDim.x + threadIdx.x;
  if (t >= (Ecnt + Nn) * 4) return;
  int eid = t >> 2, hh = t & 3, s, d;
  edge_sd(ei, eid, s, d);
  float e = edge_e(asrc, adst, s, d, hh);
  atomicAdd(&den[d * 4 + hh], expf(e - ord2f(mkey[d * 4 + hh])));
}

// one wave per (edge, head): 32 lanes cover the 64 channels (coalesced)
__global__ void k_edge_agg(const int* __restrict__ ei, const float* __restrict__ asrc,
                           const float* __restrict__ adst, const unsigned* __restrict__ mkey,
                           const float* __restrict__ den, const float* __restrict__ hbuf,
                           float* __restrict__ outacc) {
  long long t = (long long)blockIdx.x * blockDim.x + threadIdx.x;
  int lane = (int)(t & 31);
  long long pair = t >> 5;
  if (pair >= (long long)(Ecnt + Nn) * 4) return;
  int eid = (int)(pair >> 2), hh = (int)(pair & 3), s, d;
  edge_sd(ei, eid, s, d);
  float e = edge_e(asrc, adst, s, d, hh);
  float alpha = expf(e - ord2f(mkey[d * 4 + hh])) / (den[d * 4 + hh] + 1e-16f);
  const float* hp = hbuf + (size_t)s * 256 + hh * 64;
  float* op = outacc + (size_t)d * 256 + hh * 64;
  atomicAdd(&op[lane],      hp[lane]      * alpha);
  atomicAdd(&op[lane + 32], hp[lane + 32] * alpha);
}

// z_all[n, mp, c] = elu( mean_h(outacc[n,h,c]) + bias[c] )
__global__ void k_finalize(const float* __restrict__ outacc, const float* __restrict__ bias,
                           float* __restrict__ z_all, int mp) {
  int t = blockIdx.x * blockDim.x + threadIdx.x;
  if (t >= Nn * 64) return;
  int n = t >> 6, c = t & 63;
  const float* op = outacc + (size_t)n * 256 + c;
  float v = 0.25f * (op[0] + op[64] + op[128] + op[192]) + bias[c];
  v = v > 0.f ? v : expm1f(v);
  z_all[(size_t)n * 192 + mp * 64 + c] = v;
}

// per-node semantic softmax + combine (layer 2: + L2 normalize + beta output)
template<int STORE_BETA, int NORMALIZE>
__global__ void k_sem_combine(const float* __restrict__ z_all, const float* __restrict__ w_lin,
                              const float* __restrict__ sem_q, float* __restrict__ out,
                              float* __restrict__ beta_out) {
  int n = blockIdx.x * blockDim.x + threadIdx.x;
  if (n >= Nn) return;
  float sc[3];
  #pragma unroll
  for (int i = 0; i < 3; ++i) {
    const float* wp = w_lin + (size_t)n * 192 + i * 64;
    float a = 0.f;
    for (int c = 0; c < 64; ++c) a += wp[c] * sem_q[c];
    sc[i] = a;
  }
  float mx = fmaxf(sc[0], fmaxf(sc[1], sc[2]));
  float e0 = expf(sc[0] - mx), e1 = expf(sc[1] - mx), e2 = expf(sc[2] - mx);
  float inv = 1.f / (e0 + e1 + e2);
  float b0 = e0 * inv, b1 = e1 * inv, b2 = e2 * inv;
  if (STORE_BETA) { beta_out[n * 3] = b0; beta_out[n * 3 + 1] = b1; beta_out[n * 3 + 2] = b2; }
  const float* zp = z_all + (size_t)n * 192;
  float row[64];
  float nrm = 0.f;
  #pragma unroll 8
  for (int c = 0; c < 64; ++c) {
    float v = b0 * zp[c] + b1 * zp[64 + c] + b2 * zp[128 + c];
    row[c] = v; nrm += v * v;
  }
  float scale = NORMALIZE ? (1.f / fmaxf(sqrtf(nrm), 1e-12f)) : 1.f;
  for (int c = 0; c < 64; ++c) out[(size_t)n * 64 + c] = row[c] * scale;
}

// ---------------- host-side orchestration ----------------------------------
static inline int cdiv(long long a, long long b) { return (int)((a + b - 1) / b); }

static void launch_gemm(int act, const float* A, const float* B, const float* bias,
                        float* C, int M, int K, int Nc, hipStream_t s) {
  long long waves = (long long)((M + 15) / 16) * (Nc / 64);
  int blocks = cdiv(waves, 8);           // 8 waves / 256-thread block
  if (act == ACT_ELU)       k_gemm_wmma<ACT_ELU> <<<blocks, 256, 0, s>>>(A, B, bias, C, M, K, Nc);
  else if (act == ACT_TANH) k_gemm_wmma<ACT_TANH><<<blocks, 256, 0, s>>>(A, B, bias, C, M, K, Nc);
  else                      k_gemm_wmma<ACT_NONE><<<blocks, 256, 0, s>>>(A, B, bias, C, M, K, Nc);
}

struct GatP { const float* W; const float* att_dst; const float* att_src; const float* bias; };

static void run_han_layer(const float* xin, int K, const int* const ei[3], const GatP gp[3],
                          const float* semW, const float* semb, const float* semq,
                          float* hbuf, float* outacc, float* zall, float* asrc, float* adst,
                          unsigned* mkey, float* den, float* semWT,
                          float* out_z, float* beta_out, int layer2, hipStream_t s) {
  const int tot = (Ecnt + Nn) * 4;
  for (int mp = 0; mp < 3; ++mp) {
    launch_gemm(ACT_NONE, xin, gp[mp].W, nullptr, hbuf, Nn, K, 256, s);
    k_zero<<<cdiv(Nn * 4, 256), 256, 0, s>>>((float*)mkey, (size_t)Nn * 4);
    k_zero<<<cdiv(Nn * 4, 256), 256, 0, s>>>(den, (size_t)Nn * 4);
    k_zero<<<cdiv((long long)Nn * 256, 256), 256, 0, s>>>(outacc, (size_t)Nn * 256);
    k_attn_proj<<<cdiv(Nn * 4, 256), 256, 0, s>>>(hbuf, gp[mp].att_src, gp[mp].att_dst, asrc, adst);
    k_edge_max<<<cdiv(tot, 256), 256, 0, s>>>(ei[mp], asrc, adst, mkey);
    k_edge_den<<<cdiv(tot, 256), 256, 0, s>>>(ei[mp], asrc, adst, mkey, den);
    k_edge_agg<<<cdiv((long long)tot * 32, 256), 256, 0, s>>>(ei[mp], asrc, adst, mkey, den, hbuf, outacc);
    k_finalize<<<cdiv(Nn * 64, 256), 256, 0, s>>>(outacc, gp[mp].bias, zall, mp);
  }
  k_transpose64<<<16, 256, 0, s>>>(semW, semWT);
  // w_lin = tanh(z_all @ sem_W^T + sem_b), rows = N*3 (reuses hbuf)
  launch_gemm(ACT_TANH, zall, semWT, semb, hbuf, Nn * 3, 64, 64, s);
  if (layer2) k_sem_combine<1, 1><<<cdiv(Nn, 128), 128, 0, s>>>(zall, hbuf, semq, out_z, beta_out);
  else        k_sem_combine<0, 0><<<cdiv(Nn, 128), 128, 0, s>>>(zall, hbuf, semq, out_z, beta_out);
}

extern "C" void kernel_launch(void* const* d_in, const int* in_sizes, int n_in,
                              void* d_out, int out_size, void* d_ws, size_t ws_size,
                              hipStream_t stream) {
  (void)in_sizes; (void)n_in; (void)out_size; (void)ws_size;
  // jax tree flatten order: x, ei0, ei1, ei2, dec{W1,W2,b1,b2},
  // han1{mp0{W,att_dst,att_src,bias}, mp1{...}, mp2{...}, sem_W, sem_b, sem_q}, han2{...}
  const float* x = (const float*)d_in[0];
  const int* ei[3] = { (const int*)d_in[1], (const int*)d_in[2], (const int*)d_in[3] };
  const float* decW1 = (const float*)d_in[4];
  const float* decW2 = (const float*)d_in[5];
  const float* decb1 = (const float*)d_in[6];
  const float* decb2 = (const float*)d_in[7];
  GatP g1[3], g2[3];
  for (int i = 0; i < 3; ++i) {
    g1[i].W = (const float*)d_in[8 + 4 * i];  g1[i].att_dst = (const float*)d_in[9 + 4 * i];
    g1[i].att_src = (const float*)d_in[10 + 4 * i]; g1[i].bias = (const float*)d_in[11 + 4 * i];
    g2[i].W = (const float*)d_in[23 + 4 * i]; g2[i].att_dst = (const float*)d_in[24 + 4 * i];
    g2[i].att_src = (const float*)d_in[25 + 4 * i]; g2[i].bias = (const float*)d_in[26 + 4 * i];
  }
  const float* sem1W = (const float*)d_in[20];
  const float* sem1b = (const float*)d_in[21];
  const float* sem1q = (const float*)d_in[22];
  const float* sem2W = (const float*)d_in[35];
  const float* sem2b = (const float*)d_in[36];
  const float* sem2q = (const float*)d_in[37];

  float* ws     = (float*)d_ws;
  float* hbuf   = ws;                          // N*256 (also w_lin N*192, scratch)
  float* outacc = ws + (size_t)Nn * 256;       // N*256 (also decoder hidden)
  float* zall   = ws + (size_t)Nn * 512;       // N*192
  float* h1     = ws + (size_t)Nn * 704;       // N*64
  float* asrc   = ws + (size_t)Nn * 768;       // N*4
  float* adst   = ws + (size_t)Nn * 772;       // N*4
  unsigned* mkey = (unsigned*)(ws + (size_t)Nn * 776); // N*4
  float* den    = ws + (size_t)Nn * 780;       // N*4
  float* semWT  = ws + (size_t)Nn * 784;       // 64*64

  float* z_out = (float*)d_out;                // [N,64]
  float* xhat  = z_out + (size_t)Nn * 64;      // [N,128]
  float* beta  = z_out + (size_t)Nn * 192;     // [N,3]

  run_han_layer(x, INDIM, ei, g1, sem1W, sem1b, sem1q,
                hbuf, outacc, zall, asrc, adst, mkey, den, semWT, h1, nullptr, 0, stream);
  run_han_layer(h1, 64, ei, g2, sem2W, sem2b, sem2q,
                hbuf, outacc, zall, asrc, adst, mkey, den, semWT, z_out, beta, 1, stream);

  // decoder: x_hat = elu(z @ W1 + b1) @ W2 + b2
  launch_gemm(ACT_ELU,  z_out, decW1, decb1, outacc, Nn, 64, 64, stream);
  launch_gemm(ACT_NONE, outacc, decW2, decb2, xhat, Nn, 64, 128, stream);
}